// SAGE_55602646614064
// MI455X (gfx1250) — compile-verified
//
#include <hip/hip_runtime.h>

#define NN 100000   // nodes
#define NE 1600000  // edges
#define DF 64       // D_IN == H == 64

typedef __attribute__((ext_vector_type(2))) float v2f;
typedef __attribute__((ext_vector_type(8))) float v8f;

#define KP 32        // number of k-pairs (DF/2)
#define WS 80        // v2f stride per k-pair row: 640B -> adjacent k-pairs land on
                     // disjoint bank halves (bank-conflict-free across half-waves)

// ---------------------------------------------------------------- utilities
__global__ void zero_f32(float* __restrict__ p, long n) {
    long i = (long)blockIdx.x * blockDim.x + threadIdx.x;
    long stride = (long)gridDim.x * blockDim.x;
    for (; i < n; i += stride) p[i] = 0.0f;
}

__global__ void degree_kernel(const long long* __restrict__ dst,
                              float* __restrict__ deg) {
    long i = (long)blockIdx.x * blockDim.x + threadIdx.x;
    long stride = (long)gridDim.x * blockDim.x;
    for (; i < NE; i += stride)
        atomicAdd(&deg[(int)dst[i]], 1.0f);
}

__global__ void invdeg_kernel(const float* __restrict__ deg,
                              float* __restrict__ inv) {
    int i = blockIdx.x * blockDim.x + threadIdx.x;
    if (i < NN) {
        float d = deg[i];
        inv[i] = (d > 0.0f) ? (1.0f / d) : 0.0f;
    }
}

// ------------------------------------------------- scatter-add (mean aggregate)
// Feature-major: 64 consecutive threads own one edge. Coalesced 256B gathers of
// h[src]; the wave's 32 consecutive global_atomic_add_f32 coalesce in L2 (whole
// working set is L2-resident on the 192MB MI455X L2). Edge stream prefetched.
__global__ void scatter_kernel(const long long* __restrict__ src,
                               const long long* __restrict__ dst,
                               const float* __restrict__ h,
                               float* __restrict__ agg) {
    long t = (long)blockIdx.x * blockDim.x + threadIdx.x;
    long nth = (long)gridDim.x * blockDim.x;
    const int f = (int)(t & 63);
    const long estride = nth >> 6;
    for (long e = t >> 6; e < NE; e += estride) {
        __builtin_prefetch(&src[e + estride], 0, 1);
        __builtin_prefetch(&dst[e + estride], 0, 1);
        const int s = (int)src[e];
        const int d = (int)dst[e];
        atomicAdd(&agg[(long)d * DF + f], h[(long)s * DF + f]);
    }
}

// ------------------------------------------------- fused SAGE layer via WMMA
// out[row,:] = relu( (agg[row,:]*inv_deg[row]) @ Wl + bl + x[row,:] @ Wr )
// One wave computes a 16-row x 64-col tile: 4 v8f accumulators, K=64 in steps
// of 4 via V_WMMA_F32_16X16X4_F32 (full fp32 precision).
//
// Weights are staged in LDS as k-PAIRED v2f tiles: sW[kp*WS + n] =
// {W[2kp][n], W[2kp+1][n]} so every B fragment is ONE aligned ds_load_b64
// landing directly in an even VGPR pair (no v_mov repacking).
__global__ void __launch_bounds__(256)
sage_wmma_kernel(const float* __restrict__ agg,
                 const float* __restrict__ invdeg,
                 const float* __restrict__ x,
                 const float* __restrict__ Wl,
                 const float* __restrict__ bl,
                 const float* __restrict__ Wr,
                 float* __restrict__ out) {
    __shared__ v2f   sWl[KP * WS];
    __shared__ v2f   sWr[KP * WS];
    __shared__ float sB[64];

    for (int idx = threadIdx.x; idx < DF * DF; idx += 256) {
        const int kk = idx >> 6, nn = idx & 63;          // W is [k][n] row-major
        const int p  = (kk >> 1) * WS + nn;
        ((float*)&sWl[p])[kk & 1] = Wl[idx];
        ((float*)&sWr[p])[kk & 1] = Wr[idx];
    }
    if (threadIdx.x < 64) sB[threadIdx.x] = bl[threadIdx.x];
    __syncthreads();

    const int lane  = threadIdx.x & 31;
    const int wave  = threadIdx.x >> 5;
    const long rowTile = ((long)blockIdx.x * 8 + wave) * 16;
    if (rowTile >= NN) return;   // NN % 16 == 0, tiles always full

    const int mrow  = lane & 15;
    const int khalf = lane >> 4;
    const long row  = rowTile + mrow;
    const float idg = invdeg[row];
    const float* aggRow = agg + row * DF;
    const float* xRow   = x   + row * DF;

    v8f acc[4] = {};

    for (int k = 0; k < DF; k += 4) {
        const int kk = k + 2 * khalf;       // even
        const int kp = kk >> 1;             // k-pair row in LDS
        v2f aA = *(const v2f*)(aggRow + kk);
        aA.x *= idg;
        aA.y *= idg;
        const v2f aX = *(const v2f*)(xRow + kk);
#pragma unroll
        for (int t = 0; t < 4; ++t) {
            const int n = t * 16 + mrow;
            const v2f bL = sWl[kp * WS + n];   // single ds_load_b64
            const v2f bR = sWr[kp * WS + n];   // single ds_load_b64
            acc[t] = __builtin_amdgcn_wmma_f32_16x16x4_f32(
                false, aA, false, bL, (short)0, acc[t], false, false);
            acc[t] = __builtin_amdgcn_wmma_f32_16x16x4_f32(
                false, aX, false, bR, (short)0, acc[t], false, false);
        }
    }

#pragma unroll
    for (int t = 0; t < 4; ++t) {
        const float b = sB[t * 16 + mrow];
#pragma unroll
        for (int r = 0; r < 8; ++r) {
            float v = acc[t][r] + b;
            v = (v > 0.0f) ? v : 0.0f;
            out[(rowTile + r + 8 * khalf) * DF + t * 16 + mrow] = v;
        }
    }
}

// ------------------------------------------------- regression head: h @ Wfc + b
__global__ void head_kernel(const float* __restrict__ h,
                            const float* __restrict__ Wfc,
                            const float* __restrict__ bfc,
                            float* __restrict__ out) {
    __shared__ float w[64];
    if (threadIdx.x < 64) w[threadIdx.x] = Wfc[threadIdx.x];
    __syncthreads();
    const int i = blockIdx.x * blockDim.x + threadIdx.x;
    if (i >= NN) return;
    const float* hp = h + (long)i * DF;
    float s = bfc[0];
#pragma unroll
    for (int k = 0; k < DF; ++k) s += hp[k] * w[k];
    out[i] = s;
}

// ---------------------------------------------------------------- launcher
extern "C" void kernel_launch(void* const* d_in, const int* in_sizes, int n_in,
                              void* d_out, int out_size, void* d_ws, size_t ws_size,
                              hipStream_t stream) {
    const float*     x   = (const float*)d_in[0];
    const long long* ei  = (const long long*)d_in[1];   // int64 [2, NE]
    const float*     Wl1 = (const float*)d_in[2];
    const float*     bl1 = (const float*)d_in[3];
    const float*     Wr1 = (const float*)d_in[4];
    const float*     Wl2 = (const float*)d_in[5];
    const float*     bl2 = (const float*)d_in[6];
    const float*     Wr2 = (const float*)d_in[7];
    const float*     Wfc = (const float*)d_in[8];
    const float*     bfc = (const float*)d_in[9];
    float* out = (float*)d_out;

    const long long* src = ei;
    const long long* dst = ei + NE;

    // workspace carve-up (floats): deg | invdeg | agg | h1 | h2  (~78 MB)
    float* ws     = (float*)d_ws;
    float* deg    = ws;
    float* invdeg = deg + NN;
    float* agg    = invdeg + NN;
    float* h1     = agg + (long)NN * DF;
    float* h2     = h1  + (long)NN * DF;

    const dim3 blk(256);
    const int gemmGrid = (NN / 16 + 7) / 8;   // 8 waves/block, 16 rows/wave

    // ---- layer 1
    zero_f32<<<2048, blk, 0, stream>>>(deg, NN);
    zero_f32<<<2048, blk, 0, stream>>>(agg, (long)NN * DF);
    degree_kernel<<<(NE + 255) / 256, blk, 0, stream>>>(dst, deg);
    invdeg_kernel<<<(NN + 255) / 256, blk, 0, stream>>>(deg, invdeg);
    scatter_kernel<<<4096, blk, 0, stream>>>(src, dst, x, agg);
    sage_wmma_kernel<<<gemmGrid, blk, 0, stream>>>(agg, invdeg, x, Wl1, bl1, Wr1, h1);

    // ---- layer 2
    zero_f32<<<2048, blk, 0, stream>>>(agg, (long)NN * DF);
    scatter_kernel<<<4096, blk, 0, stream>>>(src, dst, h1, agg);
    sage_wmma_kernel<<<gemmGrid, blk, 0, stream>>>(agg, invdeg, h1, Wl2, bl2, Wr2, h2);

    // ---- head
    head_kernel<<<(NN + 255) / 256, blk, 0, stream>>>(h2, Wfc, bfc, out);
}